// GATLayer_35811437314194
// MI455X (gfx1250) — compile-verified
//
#include <hip/hip_runtime.h>

#define IN_C   128
#define OUT_C  64
#define HEADS  4
#define HC     256          // HEADS * OUT_C
#define NEG_SLOPE 0.2f

typedef __attribute__((ext_vector_type(2))) float v2f;
typedef __attribute__((ext_vector_type(4))) float v4f;
typedef __attribute__((ext_vector_type(8))) float v8f;
typedef __attribute__((ext_vector_type(4))) unsigned int v4u;
typedef __attribute__((ext_vector_type(8))) int v8i;
typedef __attribute__((ext_vector_type(4))) int v4i;

// ---------------------------------------------------------------------------
// Phase 1: xh[N,256] = x[N,128] @ lin_w[128,256] via V_WMMA_F32_16X16X4_F32.
//
// Block = 128 threads (4 waves). Block (bx,by) covers rows [bx*64, +64) and
// cols [by*64, +64). The 128x64 f32 weight strip (32KB) is DMA'd into LDS
// once per block by the Tensor Data Mover; each wave then computes a 16x64
// tile with 4 accumulators, streaming B fragments from LDS.
// ---------------------------------------------------------------------------
__global__ __launch_bounds__(128) void gat_gemm_wmma(
    const float* __restrict__ x, const float* __restrict__ w,
    float* __restrict__ xh, int nnodes) {
  __shared__ float lds_w[IN_C * 64];        // [k][col] strip, 32KB

  const int lane  = threadIdx.x & 31;
  const int wave  = threadIdx.x >> 5;       // 0..3 -> M sub-tile
  const int nbase = blockIdx.y * 64;        // N strip of 64
  const int row0  = blockIdx.x * 64 + wave * 16;

#if __has_builtin(__builtin_amdgcn_tensor_load_to_lds) && __has_builtin(__builtin_amdgcn_s_wait_tensorcnt)
  if (wave == 0) {
    // D# group 0: count=1, lds_addr, global_addr(tile start), type=2
    const unsigned long long gaddr =
        (unsigned long long)(uintptr_t)(w + nbase);
    const unsigned int lds_base = (unsigned int)(uintptr_t)(&lds_w[0]);
    v4u g0;
    g0.x = 1u;                                              // count=1, user mode
    g0.y = lds_base;                                        // lds_addr (bytes)
    g0.z = (unsigned int)gaddr;                             // global_addr[31:0]
    g0.w = ((unsigned int)(gaddr >> 32) & 0x01FFFFFFu)      // global_addr[56:32]
           | (2u << 30);                                    // type=2 ("image")
    // D# group 1: data_size=4B, tensor_dim0=256, tensor_dim1=128,
    //             tile_dim0=64, tile_dim1=128, tensor_dim0_stride=256
    v8i g1;
    g1[0] = (2 << 16);                                      // data_size=2 (4B)
    g1[1] = (int)(256u << 16);                              // tensor_dim0 lo16 @ bits 63:48
    g1[2] = (int)(128u << 16);                              // dim0 hi16=0 | tensor_dim1 lo16
    g1[3] = (int)(64u << 16);                               // dim1 hi16=0 | tile_dim0=64
    g1[4] = 128;                                            // tile_dim1=128, tile_dim2=0
    g1[5] = 256;                                            // tensor_dim0_stride lo32
    g1[6] = 0;                                              // stride hi16 | dim1_stride lo
    g1[7] = 0;
    v4i gz4 = {0, 0, 0, 0};                                 // 2-D tensor: groups 2/3 unused
    v8i gz8 = {0, 0, 0, 0, 0, 0, 0, 0};
    __builtin_amdgcn_tensor_load_to_lds(g0, g1, gz4, gz4, gz8, 0);
    __builtin_amdgcn_s_wait_tensorcnt(0);
  }
  __syncthreads();
#else
  // Fallback: cooperative copy of the strip into LDS.
  for (int i = threadIdx.x; i < IN_C * 64; i += 128) {
    lds_w[i] = w[(size_t)(i >> 6) * HC + nbase + (i & 63)];
  }
  __syncthreads();
#endif

  if (row0 >= nnodes) return;

  const int m    = lane & 15;               // A row / B,D column within tile
  const int half = lane >> 4;               // 0 or 1 -> which K pair

  // clamp A row for ragged tail (50000 = 781.25*64, last block is ragged)
  const int arow_i = (row0 + m < nnodes) ? (row0 + m) : (nnodes - 1);
  const float* __restrict__ arow = x + (size_t)arow_i * IN_C;

  v8f acc0 = {}, acc1 = {}, acc2 = {}, acc3 = {};

  for (int kb = 0; kb < IN_C; kb += 4) {
    const int k = kb + half * 2;
    // A 16x4: lanes 0-15 hold K={kb,kb+1}, lanes 16-31 hold K={kb+2,kb+3}
    v2f a = *(const v2f*)(arow + k);
    // B 4x16 per 16-col sub-tile from LDS: b.x = w[k][n], b.y = w[k+1][n]
    const float* __restrict__ l0 = lds_w + k * 64 + m;
    const float* __restrict__ l1 = l0 + 64;
    v2f b0 = { l0[0],  l1[0]  };
    v2f b1 = { l0[16], l1[16] };
    v2f b2 = { l0[32], l1[32] };
    v2f b3 = { l0[48], l1[48] };
    acc0 = __builtin_amdgcn_wmma_f32_16x16x4_f32(false, a, false, b0, (short)0, acc0, false, false);
    acc1 = __builtin_amdgcn_wmma_f32_16x16x4_f32(false, a, false, b1, (short)0, acc1, false, false);
    acc2 = __builtin_amdgcn_wmma_f32_16x16x4_f32(false, a, false, b2, (short)0, acc2, false, false);
    acc3 = __builtin_amdgcn_wmma_f32_16x16x4_f32(false, a, false, b3, (short)0, acc3, false, false);
  }

  // D layout: VGPR r -> row (r + 8*half), col = lane&15
  float* __restrict__ outp = xh + (size_t)row0 * HC + nbase + m;
  #pragma unroll
  for (int r = 0; r < 8; ++r) {
    const int rr = r + half * 8;
    if (row0 + rr < nnodes) {
      outp[(size_t)rr * HC +  0] = acc0[r];
      outp[(size_t)rr * HC + 16] = acc1[r];
      outp[(size_t)rr * HC + 32] = acc2[r];
      outp[(size_t)rr * HC + 48] = acc3[r];
    }
  }
}

// ---------------------------------------------------------------------------
// Phase 2: per-(node,head) attention logits.
// ---------------------------------------------------------------------------
__global__ void gat_logits(const float* __restrict__ xh,
                           const float* __restrict__ att_src,
                           const float* __restrict__ att_dst,
                           float* __restrict__ a_src, float* __restrict__ a_dst,
                           int nnodes) {
  const int idx = blockIdx.x * blockDim.x + threadIdx.x;   // (n,h)
  if (idx >= nnodes * HEADS) return;
  const int n = idx >> 2, h = idx & 3;
  const float* __restrict__ v  = xh + (size_t)n * HC + h * OUT_C;
  const float* __restrict__ as = att_src + h * OUT_C;
  const float* __restrict__ ad = att_dst + h * OUT_C;
  float ss = 0.f, sd = 0.f;
  #pragma unroll
  for (int c = 0; c < OUT_C; c += 4) {
    v4f xv = *(const v4f*)(v + c);
    v4f sv = *(const v4f*)(as + c);
    v4f dv = *(const v4f*)(ad + c);
    ss += xv.x * sv.x + xv.y * sv.y + xv.z * sv.z + xv.w * sv.w;
    sd += xv.x * dv.x + xv.y * dv.y + xv.z * dv.z + xv.w * dv.w;
  }
  a_src[idx] = ss;
  a_dst[idx] = sd;
}

// ---------------------------------------------------------------------------
// Init accumulators (d_ws is poisoned by the harness).
// ---------------------------------------------------------------------------
__global__ void gat_init(float* __restrict__ acc, float* __restrict__ denom,
                         int* __restrict__ amax, int accN, int nhN) {
  const int i = blockIdx.x * blockDim.x + threadIdx.x;
  if (i < accN) acc[i] = 0.f;
  if (i < nhN) { denom[i] = 0.f; amax[i] = (int)0x80000000; }
}

// Monotonic signed-int encoding of float for atomicMax (involution).
__device__ __forceinline__ int f2ord(float f) {
  int b = __float_as_int(f);
  return b >= 0 ? b : (b ^ 0x7fffffff);
}
__device__ __forceinline__ float ord2f(int b) {
  return __int_as_float(b >= 0 ? b : (b ^ 0x7fffffff));
}

__device__ __forceinline__ void edge_nodes(int e, int nedges,
                                           const int* __restrict__ src,
                                           const int* __restrict__ dst,
                                           int& s, int& d) {
  if (e < nedges) { s = src[e]; d = dst[e]; }
  else            { s = e - nedges; d = s; }          // self-loop
}

// ---------------------------------------------------------------------------
// Phase 3a: segment max of leaky_relu logits over incoming edges.
// ---------------------------------------------------------------------------
__global__ void gat_edge_max(const int* __restrict__ src, const int* __restrict__ dst,
                             const float* __restrict__ a_src, const float* __restrict__ a_dst,
                             int* __restrict__ amax, int nedges, int nnodes) {
  const int idx = blockIdx.x * blockDim.x + threadIdx.x;   // (e,h)
  const int tot = (nedges + nnodes) * HEADS;
  if (idx >= tot) return;
  const int e = idx >> 2, h = idx & 3;
  int s, d; edge_nodes(e, nedges, src, dst, s, d);
  float v = a_src[s * HEADS + h] + a_dst[d * HEADS + h];
  v = v > 0.f ? v : NEG_SLOPE * v;
  atomicMax(&amax[d * HEADS + h], f2ord(v));
}

// ---------------------------------------------------------------------------
// Phase 3b: exp(alpha - max), store per-edge weight, accumulate denominator.
// ---------------------------------------------------------------------------
__global__ void gat_edge_exp(const int* __restrict__ src, const int* __restrict__ dst,
                             const float* __restrict__ a_src, const float* __restrict__ a_dst,
                             const int* __restrict__ amax, float* __restrict__ ealpha,
                             float* __restrict__ denom, int nedges, int nnodes) {
  const int idx = blockIdx.x * blockDim.x + threadIdx.x;   // (e,h)
  const int tot = (nedges + nnodes) * HEADS;
  if (idx >= tot) return;
  const int e = idx >> 2, h = idx & 3;
  int s, d; edge_nodes(e, nedges, src, dst, s, d);
  float v = a_src[s * HEADS + h] + a_dst[d * HEADS + h];
  v = v > 0.f ? v : NEG_SLOPE * v;
  const float w = __expf(v - ord2f(amax[d * HEADS + h]));
  ealpha[idx] = w;
  atomicAdd(&denom[d * HEADS + h], w);
}

// ---------------------------------------------------------------------------
// Phase 3c: weighted message scatter-add. One 256-thread block per edge;
// thread c covers channel c (head = c/64). Coalesced gather + f32 atomics,
// resident in the 192MB L2 (xh + acc = ~102MB).
// ---------------------------------------------------------------------------
__global__ __launch_bounds__(HC) void gat_scatter(
    const int* __restrict__ src, const int* __restrict__ dst,
    const float* __restrict__ xh, const float* __restrict__ ealpha,
    const float* __restrict__ denom, float* __restrict__ acc,
    int nedges, int nnodes) {
  const int e = blockIdx.x;
  const int c = threadIdx.x;          // 0..255
  const int h = c >> 6;
  int s, d; edge_nodes(e, nedges, src, dst, s, d);
  const float coef = ealpha[e * HEADS + h] / (denom[d * HEADS + h] + 1e-16f);
  atomicAdd(&acc[(size_t)d * HC + c], xh[(size_t)s * HC + c] * coef);
}

// ---------------------------------------------------------------------------
// Phase 4: mean over heads + biases + relu.
// ---------------------------------------------------------------------------
__global__ void gat_final(const float* __restrict__ acc,
                          const float* __restrict__ cbias,
                          const float* __restrict__ lbias,
                          float* __restrict__ out, int nnodes) {
  const int idx = blockIdx.x * blockDim.x + threadIdx.x;   // (n,c)
  if (idx >= nnodes * OUT_C) return;
  const int n = idx >> 6, c = idx & 63;
  const float* __restrict__ a = acc + (size_t)n * HC + c;
  float v = (a[0] + a[64] + a[128] + a[192]) * 0.25f + cbias[c] + lbias[c];
  out[idx] = v > 0.f ? v : 0.f;
}

extern "C" void kernel_launch(void* const* d_in, const int* in_sizes, int n_in,
                              void* d_out, int out_size, void* d_ws, size_t ws_size,
                              hipStream_t stream) {
  const float* x          = (const float*)d_in[0];
  const int*   eidx       = (const int*)  d_in[1];
  const float* lin_w      = (const float*)d_in[2];
  const float* att_src    = (const float*)d_in[3];
  const float* att_dst    = (const float*)d_in[4];
  const float* conv_bias  = (const float*)d_in[5];
  const float* layer_bias = (const float*)d_in[6];

  const int nnodes = in_sizes[0] / IN_C;
  const int nedges = in_sizes[1] / 2;
  const int* src = eidx;
  const int* dst = eidx + nedges;

  // workspace carve-up
  float* ws = (float*)d_ws;
  size_t off = 0;
  float* xh     = ws + off; off += (size_t)nnodes * HC;
  float* acc    = ws + off; off += (size_t)nnodes * HC;
  float* a_src  = ws + off; off += (size_t)nnodes * HEADS;
  float* a_dst  = ws + off; off += (size_t)nnodes * HEADS;
  float* denom  = ws + off; off += (size_t)nnodes * HEADS;
  float* ealpha = ws + off; off += (size_t)(nedges + nnodes) * HEADS;
  int*   amax   = (int*)(ws + off);

  const int etot = nedges + nnodes;
  const int nh   = nnodes * HEADS;

  dim3 ggrid((nnodes + 63) / 64, HC / 64);   // 64-row groups x 4 column strips
  gat_gemm_wmma<<<ggrid, 128, 0, stream>>>(x, lin_w, xh, nnodes);
  gat_logits<<<(nh + 255) / 256, 256, 0, stream>>>(xh, att_src, att_dst, a_src, a_dst, nnodes);
  gat_init<<<((nnodes * HC) + 255) / 256, 256, 0, stream>>>(acc, denom, amax, nnodes * HC, nh);
  gat_edge_max<<<(etot * HEADS + 255) / 256, 256, 0, stream>>>(src, dst, a_src, a_dst, amax, nedges, nnodes);
  gat_edge_exp<<<(etot * HEADS + 255) / 256, 256, 0, stream>>>(src, dst, a_src, a_dst, amax, ealpha, denom, nedges, nnodes);
  gat_scatter<<<etot, HC, 0, stream>>>(src, dst, xh, ealpha, denom, acc, nedges, nnodes);
  gat_final<<<((nnodes * OUT_C) + 255) / 256, 256, 0, stream>>>(acc, conv_bias, layer_bias, (float*)d_out, nnodes);
}